// SENGR_GCN_50319836840483
// MI455X (gfx1250) — compile-verified
//
#include <hip/hip_runtime.h>

#define NUM_USERS 50000
#define NUM_ITEMS 50000
#define N_NODES   100000   // NUM_USERS + NUM_ITEMS
#define DIM       64
#define NUM_EDGES 3200000

typedef __attribute__((ext_vector_type(2))) float v2f;
typedef __attribute__((ext_vector_type(8))) float v8f;

// ---------------------------------------------------------------------------
// Zero the aggregation buffer (graph-capture safe, no hipMemsetAsync needed).
// ---------------------------------------------------------------------------
__global__ __launch_bounds__(256) void gcn_zero_f4(float4* __restrict__ p, int n4) {
    int i = blockIdx.x * 256 + threadIdx.x;
    if (i < n4) p[i] = make_float4(0.f, 0.f, 0.f, 0.f);
}

// ---------------------------------------------------------------------------
// Scatter phase: for each edge e, agg[dst[e]] += x[src[e]] * w[e].
// 16 threads per edge, float4 per thread -> 256B contiguous per edge row.
// x rows (25.6 MB) and edges (38.4 MB) are L2-resident on MI455X (192 MB L2),
// so this is bound by L2 f32-atomic throughput; use HW global_atomic_add_f32.
// split: rows < split come from xa, rows >= split from xb (layer1: user/item).
// ---------------------------------------------------------------------------
__global__ __launch_bounds__(256) void gcn_scatter(
    const int*   __restrict__ ei,   // [2, E]
    const float* __restrict__ ew,   // [E]
    const float* __restrict__ xa,
    const float* __restrict__ xb,
    int split,
    float* __restrict__ agg)        // [N, 64]
{
    unsigned t   = blockIdx.x * 256u + threadIdx.x;   // grid sized exactly E*16
    unsigned e   = t >> 4;
    unsigned seg = (t & 15u) << 2;                    // float4 segment within row

    int   src = ei[e];
    int   dst = ei[NUM_EDGES + e];
    float w   = ew[e];

    const float* xr = (src < split) ? (xa + (size_t)src * DIM)
                                    : (xb + (size_t)(src - split) * DIM);
    float4 v = *(const float4*)(xr + seg);

    float* ar = agg + (size_t)dst * DIM + seg;
    unsafeAtomicAdd(ar + 0, v.x * w);
    unsafeAtomicAdd(ar + 1, v.y * w);
    unsafeAtomicAdd(ar + 2, v.z * w);
    unsafeAtomicAdd(ar + 3, v.w * w);
}

// ---------------------------------------------------------------------------
// Dense phase: out = ((agg + x) * 0.5) @ W^T + b   via V_WMMA_F32_16X16X4_F32.
// Block = 256 threads = 8 waves, covers 32 node rows.
//   upd[32][64] staged in LDS once (row stride 66 floats to spread banks).
//   Wave w: node-tile = w>>2 (16 rows), out-tile = w&3 (16 output dims).
//   K=64 -> 16 chained f32 WMMAs accumulating in an 8-VGPR C fragment.
// A frag (16x4 f32): lanes 0-15 -> K = k0,k0+1 ; lanes 16-31 -> K = k0+2,k0+3
// B frag (4x16 f32): lane%16 = N, same K-halving; B[k][n] = W[outBase+n][k]
// C/D  (16x16 f32): VGPR v -> M = v + (lane<16 ? 0 : 8), N = lane%16
// ---------------------------------------------------------------------------
__global__ __launch_bounds__(256) void gcn_linear_wmma(
    const float* __restrict__ agg,   // [N, 64]
    const float* __restrict__ xa,
    const float* __restrict__ xb,
    int split,
    const float* __restrict__ W,     // [64, 64]
    const float* __restrict__ bias,  // [64]
    float* __restrict__ out)         // [N, 64]
{
    __shared__ float upd[32][DIM + 2];

    const int tid      = threadIdx.x;
    const int nodeBase = blockIdx.x * 32;

    // Stage (agg + x) * 0.5 for this block's 32 rows (2048 floats, 8 per thread).
    #pragma unroll
    for (int k = 0; k < 8; ++k) {
        int idx = tid + k * 256;
        int row = idx >> 6;
        int col = idx & 63;
        int g   = nodeBase + row;
        float xv = (g < split) ? xa[(size_t)g * DIM + col]
                               : xb[(size_t)(g - split) * DIM + col];
        upd[row][col] = (agg[(size_t)g * DIM + col] + xv) * 0.5f;
    }
    __syncthreads();

    const int lane    = tid & 31;
    const int wave    = tid >> 5;
    const int n       = lane & 15;                 // output column within tile
    const int khalf   = (lane < 16) ? 0 : 2;       // K offset for upper lane half
    const int rowLoc  = ((wave >> 2) << 4) + n;    // A-matrix row (M) in LDS
    const int outBase = (wave & 3) << 4;           // output-dim tile base

    const float* wrow = W + (size_t)(outBase + n) * DIM;  // B[k][n] = W[outBase+n][k]

    v8f c = {};
    #pragma unroll
    for (int k0 = 0; k0 < DIM; k0 += 4) {
        v2f a, b;
        a.x = upd[rowLoc][k0 + khalf];
        a.y = upd[rowLoc][k0 + khalf + 1];
        b.x = wrow[k0 + khalf];
        b.y = wrow[k0 + khalf + 1];
        c = __builtin_amdgcn_wmma_f32_16x16x4_f32(
                /*neg_a=*/false, a, /*neg_b=*/false, b,
                /*c_mod=*/(short)0, c, /*reuse_a=*/false, /*reuse_b=*/false);
    }

    const float bv       = bias[outBase + n];
    const int   tileRow0 = nodeBase + ((wave >> 2) << 4);
    const int   mOff     = (lane < 16) ? 0 : 8;
    #pragma unroll
    for (int v = 0; v < 8; ++v) {
        int M = v + mOff;
        out[(size_t)(tileRow0 + M) * DIM + outBase + n] = c[v] + bv;
    }
}

// ---------------------------------------------------------------------------
// Host-side orchestration (all launches on `stream`, capture-safe).
// Workspace: 2 x [N,64] f32 buffers = 51.2 MB.
// ---------------------------------------------------------------------------
extern "C" void kernel_launch(void* const* d_in, const int* in_sizes, int n_in,
                              void* d_out, int out_size, void* d_ws, size_t ws_size,
                              hipStream_t stream) {
    const int*   ei = (const int*)  d_in[0];   // edge_index [2, E]
    const float* ew = (const float*)d_in[1];   // edge_weight [E]
    const float* ue = (const float*)d_in[2];   // user_emb [50000, 64]
    const float* ie = (const float*)d_in[3];   // item_emb [50000, 64]
    const float* W1 = (const float*)d_in[4];
    const float* b1 = (const float*)d_in[5];
    const float* W2 = (const float*)d_in[6];
    const float* b2 = (const float*)d_in[7];
    float*       out = (float*)d_out;

    float* agg = (float*)d_ws;                        // [N, 64]
    float* x1  = agg + (size_t)N_NODES * DIM;         // [N, 64] layer-1 output

    const int n4 = N_NODES * DIM / 4;                 // 1.6M float4
    const dim3 zgrid((n4 + 255) / 256);
    const dim3 sgrid(NUM_EDGES * 16 / 256);           // 200000 blocks, exact
    const dim3 lgrid(N_NODES / 32);                   // 3125 blocks, exact

    // ---- Layer 1: x = [user_emb ; item_emb] (split pointers, no copy) ----
    gcn_zero_f4   <<<zgrid, 256, 0, stream>>>((float4*)agg, n4);
    gcn_scatter   <<<sgrid, 256, 0, stream>>>(ei, ew, ue, ie, NUM_USERS, agg);
    gcn_linear_wmma<<<lgrid, 256, 0, stream>>>(agg, ue, ie, NUM_USERS, W1, b1, x1);

    // ---- Layer 2: x = x1 (split = N -> always first pointer) ----
    gcn_zero_f4   <<<zgrid, 256, 0, stream>>>((float4*)agg, n4);
    gcn_scatter   <<<sgrid, 256, 0, stream>>>(ei, ew, x1, x1, N_NODES, agg);
    gcn_linear_wmma<<<lgrid, 256, 0, stream>>>(agg, x1, x1, N_NODES, W2, b2, out);
}